// STFNConv_89687507076371
// MI455X (gfx1250) — compile-verified
//
#include <hip/hip_runtime.h>

// GCNConv forward for MI455X (gfx1250, wave32).
//   h = x @ W              -> V_WMMA_F32_16X16X4_F32 (fp32 matrix pipe)
//   deg/dinv               -> fp32 atomic histogram + rsqrt
//   out = scatter-add(h[src]*norm) + self-loop + bias  -> global_atomic_add_f32

typedef __attribute__((ext_vector_type(2))) float v2f;
typedef __attribute__((ext_vector_type(8))) float v8f;

#define CH 64  // IN_CH == OUT_CH == 64

// ---------------- degree / normalization ----------------

__global__ void deg_init_kernel(float* __restrict__ deg, int N) {
    int i = blockIdx.x * blockDim.x + threadIdx.x;
    if (i < N) deg[i] = 1.0f;  // self-loop contributes 1 to every node's in-degree
}

__global__ void deg_accum_kernel(const int* __restrict__ eidx, float* __restrict__ deg, int E) {
    int e = blockIdx.x * blockDim.x + threadIdx.x;
    if (e < E) {
        int d = eidx[E + e];  // edge_index[1][e] (dst)
        unsafeAtomicAdd(&deg[d], 1.0f);  // exact: degrees << 2^24
    }
}

__global__ void rsqrt_kernel(float* __restrict__ deg_dinv, int N) {
    int i = blockIdx.x * blockDim.x + threadIdx.x;
    if (i < N) deg_dinv[i] = rsqrtf(deg_dinv[i]);  // deg >= 1 always (self-loop)
}

// ---------------- h = x @ W via WMMA f32 16x16x4 ----------------
// Block: 256 threads = 8 waves. Each wave computes a 16-row x 64-col strip of h.
// W (64x64 f32, 16 KB) is staged in LDS once per block.

__global__ void __launch_bounds__(256) gemm_wmma_kernel(const float* __restrict__ x,
                                                        const float* __restrict__ W,
                                                        float* __restrict__ h, int N) {
    __shared__ float Ws[CH * CH];  // 16 KB of 320 KB WGP LDS

    for (int i = threadIdx.x; i < CH * CH / 4; i += 256)
        ((float4*)Ws)[i] = ((const float4*)W)[i];
    __syncthreads();

    const int wave = threadIdx.x >> 5;
    const int lane = threadIdx.x & 31;
    const int m    = lane & 15;   // row-in-tile (A) / col-in-tile (B,C,D)
    const int hi   = lane >> 4;   // which K-half of the fragment this lane holds

    const long long row0 = (long long)blockIdx.x * 128 + (long long)wave * 16;
    if (row0 + 16 > (long long)N) return;  // wave-uniform guard: EXEC stays all-1s for WMMA

    v8f acc0 = {}, acc1 = {}, acc2 = {}, acc3 = {};  // 4 N-tiles of 16x16
    const float* xrow = x + (row0 + m) * CH;

#pragma unroll
    for (int k = 0; k < 16; ++k) {  // K = 64 in steps of 4
        // A fragment (16x4): v0 = A[m][4k + 2*hi], v1 = A[m][4k + 2*hi + 1]
        v2f a = *(const v2f*)(xrow + 4 * k + 2 * hi);

        // B fragments (4x16): v0 = W[4k+2*hi][col], v1 = W[4k+2*hi+1][col]
        const float* wb = Ws + (4 * k + 2 * hi) * CH + m;
        v2f b0 = { wb[0],  wb[CH + 0]  };
        v2f b1 = { wb[16], wb[CH + 16] };
        v2f b2 = { wb[32], wb[CH + 32] };
        v2f b3 = { wb[48], wb[CH + 48] };

        acc0 = __builtin_amdgcn_wmma_f32_16x16x4_f32(false, a, false, b0, (short)0, acc0, false, false);
        acc1 = __builtin_amdgcn_wmma_f32_16x16x4_f32(false, a, false, b1, (short)0, acc1, false, false);
        acc2 = __builtin_amdgcn_wmma_f32_16x16x4_f32(false, a, false, b2, (short)0, acc2, false, false);
        acc3 = __builtin_amdgcn_wmma_f32_16x16x4_f32(false, a, false, b3, (short)0, acc3, false, false);
    }

    // C/D layout: VGPR r holds M = r + 8*hi, N = m
#pragma unroll
    for (int r = 0; r < 8; ++r) {
        float* o = h + (row0 + r + 8 * hi) * CH + m;
        o[0]  = acc0[r];
        o[16] = acc1[r];
        o[32] = acc2[r];
        o[48] = acc3[r];
    }
}

// ---------------- out = h * dinv^2 + b (self-loop term, also initializes d_out) --------

__global__ void init_out_kernel(const float* __restrict__ h, const float* __restrict__ dinv,
                                const float* __restrict__ bias, float* __restrict__ out,
                                long long total) {
    long long i = (long long)blockIdx.x * blockDim.x + threadIdx.x;
    if (i < total) {
        int node = (int)(i >> 6);
        int c    = (int)(i & 63);
        float di = dinv[node];
        out[i] = h[i] * di * di + bias[c];
    }
}

// ---------------- edge gather-scale-scatter ----------------
// One wave per edge: 32 lanes x float2 = 64 channels, coalesced 256B per edge.

__global__ void __launch_bounds__(256) scatter_kernel(const float* __restrict__ h,
                                                      const int* __restrict__ eidx,
                                                      const float* __restrict__ dinv,
                                                      float* __restrict__ out, int E) {
    int e = blockIdx.x * (blockDim.x >> 5) + (threadIdx.x >> 5);
    if (e >= E) return;
    int lane = threadIdx.x & 31;

    int s = eidx[e];      // src
    int d = eidx[E + e];  // dst
    float nrm = dinv[s] * dinv[d];

    v2f v = *(const v2f*)(h + (long long)s * CH + 2 * lane);
    float* o = out + (long long)d * CH + 2 * lane;
    unsafeAtomicAdd(o,     v.x * nrm);  // -> global_atomic_add_f32
    unsafeAtomicAdd(o + 1, v.y * nrm);
}

// ---------------- launcher ----------------

extern "C" void kernel_launch(void* const* d_in, const int* in_sizes, int n_in,
                              void* d_out, int out_size, void* d_ws, size_t ws_size,
                              hipStream_t stream) {
    const float* x    = (const float*)d_in[0];
    const int*   eidx = (const int*)d_in[1];   // [2, E] flat: row 0 = src, row 1 = dst
    const float* W    = (const float*)d_in[2];
    const float* bias = (const float*)d_in[3];
    float*       out  = (float*)d_out;

    const int N = in_sizes[0] / CH;
    const int E = in_sizes[1] / 2;

    // workspace layout: h [N*64 f32] | dinv [N f32]
    float* h    = (float*)d_ws;
    float* dinv = h + (size_t)N * CH;

    deg_init_kernel<<<(N + 255) / 256, 256, 0, stream>>>(dinv, N);
    deg_accum_kernel<<<(E + 255) / 256, 256, 0, stream>>>(eidx, dinv, E);
    rsqrt_kernel<<<(N + 255) / 256, 256, 0, stream>>>(dinv, N);

    gemm_wmma_kernel<<<(N + 127) / 128, 256, 0, stream>>>(x, W, h, N);

    long long total = (long long)N * CH;
    init_out_kernel<<<(int)((total + 255) / 256), 256, 0, stream>>>(h, dinv, bias, out, total);

    scatter_kernel<<<(E + 7) / 8, 256, 0, stream>>>(h, eidx, dinv, out, E);
}